// CosineDist_42013370089991
// MI455X (gfx1250) — compile-verified
//
#include <hip/hip_runtime.h>
#include <hip/hip_bf16.h>
#include <math.h>

// Problem constants (fixed shapes from the reference)
#define N_POS      1024
#define EMBED_DIM  128
#define N_SEG      64
#define EPS_F      1e-8f

typedef __attribute__((ext_vector_type(2))) float v2f;
typedef __attribute__((ext_vector_type(8))) float v8f;

// ---------------------------------------------------------------------------
// Pass 1: fold segment-mean weights and 1/|t_p| into a single 128-dim vector
//   v[d] = sum_p  t_p[d] / (64 * max(count[id_p],1) * (|t_p| + eps))
// One block of 1024 threads (one thread per target row). Deterministic:
// integer histogram atomics + sequential per-dim reduction (no float atomics).
// ---------------------------------------------------------------------------
__global__ __launch_bounds__(1024) void cosdist_prep_kernel(
    const float* __restrict__ target,        // [N_POS, EMBED_DIM]
    const int*   __restrict__ ids,           // [N_POS]
    float*       __restrict__ v_out)         // [EMBED_DIM] in workspace
{
    __shared__ int   s_cnt[N_SEG];
    __shared__ float s_w[N_POS];

    const int tid = threadIdx.x;
    if (tid < N_SEG) s_cnt[tid] = 0;
    __syncthreads();

    const int my_id = ids[tid];
    atomicAdd(&s_cnt[my_id], 1);

    // ||t_p|| via float4 loads
    float ssq = 0.0f;
    const float4* row = reinterpret_cast<const float4*>(target + (size_t)tid * EMBED_DIM);
    #pragma unroll
    for (int i = 0; i < EMBED_DIM / 4; ++i) {
        float4 t = row[i];
        ssq += t.x * t.x + t.y * t.y + t.z * t.z + t.w * t.w;
    }
    __syncthreads();

    int   c   = s_cnt[my_id];
    float cnt = (float)(c < 1 ? 1 : c);
    s_w[tid]  = 1.0f / (64.0f * cnt * (sqrtf(ssq) + EPS_F));
    __syncthreads();

    // Sequential (deterministic) weighted reduction: one thread per dim.
    if (tid < EMBED_DIM) {
        float acc = 0.0f;
        for (int p = 0; p < N_POS; ++p)
            acc += s_w[p] * target[(size_t)p * EMBED_DIM + tid];
        v_out[tid] = acc;
    }
}

// ---------------------------------------------------------------------------
// Pass 2: out[n] = -(v . pred[n]) / (|pred[n]| + eps)      (bandwidth bound)
// One wave32 per 16-node tile. Dot products run on the matrix pipe:
//   V_WMMA_F32_16X16X4_F32, A = pred tile (16xK=4 per step),
//   B = v-chunk broadcast across all 16 columns, 32 steps over K=128.
// A-fragment layout (ISA 7.12.2, 32-bit A 16x4): lane L holds row M=L%16,
// K = {0,1} for lanes 0-15 / {2,3} for lanes 16-31  -> one 8B load per step.
// Row norms accumulate in VALU alongside; xor-16 shuffle merges halves.
// C/D layout: VGPR r, lane L -> M = r + 8*(L/16); all N columns equal.
// ---------------------------------------------------------------------------
__global__ __launch_bounds__(256) void cosdist_main_kernel(
    const float* __restrict__ pred,          // [n_nodes, EMBED_DIM]
    const float* __restrict__ v,             // [EMBED_DIM]
    float*       __restrict__ out,           // [n_nodes]
    int ntiles)                              // n_nodes / 16
{
    const int wave = (int)((blockIdx.x * blockDim.x + threadIdx.x) >> 5);
    const int lane = (int)(threadIdx.x & 31);
    if (wave >= ntiles) return;              // wave-uniform: EXEC all-1s inside

    const int base = wave * 16;
    const int m    = lane & 15;              // A row this lane feeds
    const int o    = (lane >> 4) << 1;       // K sub-offset: 0 (lo half) / 2 (hi half)

    // Preload B fragments: B[k][n] = v[4*kk + k] for every column n.
    v2f b[32];
    #pragma unroll
    for (int kk = 0; kk < 32; ++kk) {
        b[kk].x = v[4 * kk + o];
        b[kk].y = v[4 * kk + o + 1];
    }

    v8f  c   = {};
    float nsq = 0.0f;
    const float* rowp = pred + (size_t)(base + m) * EMBED_DIM + o;

    #pragma unroll
    for (int kk = 0; kk < 32; ++kk) {
        v2f a = *reinterpret_cast<const v2f*>(rowp + 4 * kk);
        nsq += a.x * a.x + a.y * a.y;
        // (neg_a, A, neg_b, B, c_mod, C, reuse_a, reuse_b)
        c = __builtin_amdgcn_wmma_f32_16x16x4_f32(
                false, a, false, b[kk], (short)0, c, false, false);
    }

    // Lane L now holds partial ||row m||^2 over its K-half; merge halves.
    nsq += __shfl_xor(nsq, 16, 32);          // lane L: full ||pred[base + L%16]||^2

    // Extract results: lane 0 holds rows 0..7 in c[0..7], lane 16 rows 8..15.
    #pragma unroll
    for (int r = 0; r < 8; ++r) {
        float nlo = __shfl(nsq, r,     32);  // ||row r||^2
        float nhi = __shfl(nsq, r + 8, 32);  // ||row r+8||^2
        if (lane == 0)
            out[base + r]     = -c[r] / (sqrtf(nlo) + EPS_F);
        if (lane == 16)
            out[base + r + 8] = -c[r] / (sqrtf(nhi) + EPS_F);
    }
}

// ---------------------------------------------------------------------------
extern "C" void kernel_launch(void* const* d_in, const int* in_sizes, int n_in,
                              void* d_out, int out_size, void* d_ws, size_t ws_size,
                              hipStream_t stream) {
    const float* pred   = (const float*)d_in[0];   // [n_nodes, 128] f32
    const float* target = (const float*)d_in[1];   // [1024, 128]   f32
    const int*   ids    = (const int*)d_in[2];     // [1024]        i32
    float*       out    = (float*)d_out;           // [n_nodes]     f32
    float*       v      = (float*)d_ws;            // 128 floats of scratch

    cosdist_prep_kernel<<<1, 1024, 0, stream>>>(target, ids, v);

    const int n_nodes = out_size;                  // 100000, multiple of 16
    const int ntiles  = n_nodes / 16;              // 6250 wave-tiles
    const int waves_per_block = 256 / 32;          // 8
    const int blocks  = (ntiles + waves_per_block - 1) / waves_per_block;

    cosdist_main_kernel<<<blocks, 256, 0, stream>>>(pred, v, out, ntiles);
}